// GConv_58961311039765
// MI455X (gfx1250) — compile-verified
//
#include <hip/hip_runtime.h>
#include <hip/hip_bf16.h>

typedef float v2f __attribute__((ext_vector_type(2)));
typedef float v8f __attribute__((ext_vector_type(8)));

// Problem constants (fixed by the reference)
#define NB   16
#define CC   64
#define RR   8
#define HH   64
#define WW   64
#define OO   64
#define NKP  5
#define NROT 8

// Ring offsets packed as (d+1) in 2-bit fields, index = ring position 0..7
// dy+1: {0,0,0,1,2,2,2,1} -> 0x6A40 ; dx+1: {0,1,2,2,2,1,0,0} -> 0x06A4
#define RING_DY_PACKED 0x6A40
#define RING_DX_PACKED 0x06A4

#define XROWS 4           // two output rows -> 4-row halo window
#define XCOLS 66          // W + 2 halo
#define XPAD  68          // padded LDS row stride (floats)
#define WPAD  65          // padded o-stride for weight slab

__global__ __launch_bounds__(256)
void GConv_wmma_f32(const float* __restrict__ x,
                    const float* __restrict__ wgt,
                    float* __restrict__ out)
{
    __shared__ float Xs[CC][XROWS][XPAD]; // 64*4*68*4 = 69,632 B
    __shared__ float Ws[NKP][CC][WPAD];   // 5*64*65*4 = 83,200 B

    const int h0  = blockIdx.x * 2;       // first of two output rows
    const int nz  = blockIdx.y;           // n*8 + rot
    const int n   = nz >> 3;
    const int rot = nz & 7;

    const int tid   = threadIdx.x;
    const int lane  = tid & 31;
    const int wave  = tid >> 5;           // 0..7
    const int hrow  = wave & 1;           // which of the 2 output rows
    const int nbase = ((wave >> 1) & 1) * 32;   // 2 n-supertiles of 32 px
    const int mbase = (wave >> 2) * 32;         // 2 m-supertiles of 32 o
    const int lhalf = lane >> 4;
    const int l16   = lane & 15;

    // staging decompositions (shift-only)
    const int cg = tid >> 5;              // 0..7 : Xs channel group
    const int og = tid >> 6;              // 0..3 : Ws o group
    const int cl = tid & 63;              // Ws channel lane

    v8f acc00 = {0.f,0.f,0.f,0.f,0.f,0.f,0.f,0.f};
    v8f acc01 = acc00, acc10 = acc00, acc11 = acc00;

    for (int r = 0; r < RR; ++r) {
        const int idx = (r + rot) & 7;    // rotated x slice feeding this r
        __syncthreads();                  // previous iteration done reading LDS

        // ---- stage x window: 64 ch x 4 rows x 66 cols (zero halo), coalesced ----
        const float* xr = x + (((size_t)n * CC) * RR + idx) * (HH * WW);
        for (int c = cg; c < CC; c += 8) {
            const float* xc = xr + (size_t)c * (RR * HH * WW);
#pragma unroll
            for (int row = 0; row < XROWS; ++row) {
                const int hs = h0 + row - 1;
                const bool hok = (unsigned)hs < HH;
                const float* xrow = xc + hs * WW;
                for (int col = lane; col < XCOLS; col += 32) {
                    const int ws = col - 1;
                    float v = (hok && (unsigned)ws < WW) ? xrow[ws] : 0.f;
                    Xs[c][row][col] = v;
                }
            }
        }

        // ---- stage weight slab for this r: all 5 kernel points at once ----
        for (int o = og; o < OO; o += 4) {
            const float* wp = wgt + (((size_t)o * CC + cl) * RR + r) * NKP;
#pragma unroll
            for (int k = 0; k < NKP; ++k)
                Ws[k][cl][o] = wp[k];
        }

        // prefetch next r's x window while computing on this one
        if (r + 1 < RR) {
            const int idx2 = (r + 1 + rot) & 7;
            const float* nx = x + (((size_t)n * CC) * RR + idx2) * (HH * WW);
            const int pc  = tid >> 2;         // 0..63 channel
            const int prw = tid & 3;          // 0..3 row of halo window
            const int hs  = h0 + prw - 1;
            if ((unsigned)hs < HH) {
                const float* p = nx + (size_t)pc * (RR * HH * WW) + hs * WW;
                __builtin_prefetch(p, 0, 3);       // global_prefetch_b8
                __builtin_prefetch(p + 32, 0, 3);  // second half of the row
            }
        }
        __syncthreads();

        // ---- compute: 2 A frags + 2 B frags feed 4 WMMAs per K-step ----
#pragma unroll
        for (int k = 0; k < NKP; ++k) {
            int dy1 = 1, dx1 = 1;             // center point (d+1 form)
            if (k > 0) {
                const int rr = (2 * k - 1 + rot) & 7;
                dy1 = (RING_DY_PACKED >> (2 * rr)) & 3;
                dx1 = (RING_DX_PACKED >> (2 * rr)) & 3;
            }
            const int brow = hrow + dy1;                   // 0..3
            const int bcol = nbase + l16 + dx1;            // 0..65

            for (int c0 = 0; c0 < CC; c0 += 4) {
                const int kc = c0 + 2 * lhalf;             // K-half split per ISA layout
                v2f a0, a1, b0, b1;
                a0.x = Ws[k][kc + 0][mbase + l16];
                a0.y = Ws[k][kc + 1][mbase + l16];
                a1.x = Ws[k][kc + 0][mbase + 16 + l16];
                a1.y = Ws[k][kc + 1][mbase + 16 + l16];
                b0.x = Xs[kc + 0][brow][bcol];
                b0.y = Xs[kc + 1][brow][bcol];
                b1.x = Xs[kc + 0][brow][bcol + 16];
                b1.y = Xs[kc + 1][brow][bcol + 16];
                acc00 = __builtin_amdgcn_wmma_f32_16x16x4_f32(
                            false, a0, false, b0, (short)0, acc00, false, false);
                acc01 = __builtin_amdgcn_wmma_f32_16x16x4_f32(
                            false, a0, false, b1, (short)0, acc01, false, false);
                acc10 = __builtin_amdgcn_wmma_f32_16x16x4_f32(
                            false, a1, false, b0, (short)0, acc10, false, false);
                acc11 = __builtin_amdgcn_wmma_f32_16x16x4_f32(
                            false, a1, false, b1, (short)0, acc11, false, false);
            }
        }
    }

    // C/D 16x16 f32 layout: VGPR i -> M = i (lanes 0-15) or 8+i (lanes 16-31)
    const int h = h0 + hrow;
    float* ob = out + ((((size_t)n * OO + mbase) * NROT + rot) * HH + h) * WW
                    + nbase + l16;
    const size_t ostride = (size_t)(NROT * HH * WW);       // per +1 in o
#pragma unroll
    for (int i = 0; i < 8; ++i) {
        const size_t om = (size_t)(i + 8 * lhalf) * ostride;
        ob[om]                     = acc00[i];
        ob[om + 16]                = acc01[i];
        ob[om + 16 * ostride]      = acc10[i];
        ob[om + 16 * ostride + 16] = acc11[i];
    }
}

extern "C" void kernel_launch(void* const* d_in, const int* in_sizes, int n_in,
                              void* d_out, int out_size, void* d_ws, size_t ws_size,
                              hipStream_t stream) {
    const float* x   = (const float*)d_in[0];
    const float* wgt = (const float*)d_in[1];
    float* out       = (float*)d_out;
    dim3 grid(HH / 2, NB * NROT);   // (32, 128): two (n, rot) rows per block
    GConv_wmma_f32<<<grid, 256, 0, stream>>>(x, wgt, out);
}